// TPGNN_1254130450614
// MI455X (gfx1250) — compile-verified
//
#include <hip/hip_runtime.h>
#include <math.h>

// ---------------------------------------------------------------------------
// TPGNN on MI455X (gfx1250, wave32).
//  - GEMMs via V_WMMA_F32_16X16X4_F32 (full f32 precision; GEMM cost is tiny).
//  - Edge phase: float4 gathers + GLOBAL_ATOMIC_ADD_F32 scatter; gate computed
//    once per edge (identical across all 3 layers).
//  - All intermediates live in workspace -> L2-resident (192MB L2 >> 150MB
//    peak live set), so random gathers/atomics hit L2, not HBM.
// ---------------------------------------------------------------------------

typedef float v2f __attribute__((ext_vector_type(2)));
typedef float v8f __attribute__((ext_vector_type(8)));

#define N_NODES 100000
#define E_EDGES 1600000
#define HDIM    64
#define DOUT    153
#define DOUT_P  160   // padded to 10 WMMA column tiles / 40 float4 groups

// ---------------------------------------------------------------------------
// Zero-fill
// ---------------------------------------------------------------------------
__global__ void fill_zero_kernel(float* __restrict__ p, int n) {
    int i = blockIdx.x * blockDim.x + threadIdx.x;
    if (i < n) p[i] = 0.0f;
}

// ---------------------------------------------------------------------------
// Per-edge gate: sigmoid(clip(trust[row],0,1) * clip(pcoeff,-5,5))
// ---------------------------------------------------------------------------
__global__ void edge_gate_kernel(const int* __restrict__ row,
                                 const float* __restrict__ trust,
                                 const float* __restrict__ pcoeff,
                                 float* __restrict__ gate) {
    int e = blockIdx.x * blockDim.x + threadIdx.x;
    if (e >= E_EDGES) return;
    float t = trust[row[e]];
    t = fminf(fmaxf(t, 0.0f), 1.0f);
    float p = fminf(fmaxf(pcoeff[e], -5.0f), 5.0f);
    gate[e] = 1.0f / (1.0f + __expf(-t * p));
}

// ---------------------------------------------------------------------------
// GEMM: out[n, j] = sum_k in[n,k] * W[j,k] + bias[j]
//   in : [N_NODES x 64] row-major, W : [JMAX x 64] row-major
//   out: stride OS (JT*16 columns written; j >= JMAX columns come out as 0.0
//        because both W fragment and bias are loaded as 0 there).
// One wave per 16-row strip. K=64 -> 16 steps of V_WMMA_F32_16X16X4_F32.
// ISA fragment layouts (wave32):
//   A 16x4 f32 : lane L<16 -> M=L, {v0,v1}={K0,K1}; lane>=16 -> M=L-16, {K2,K3}
//   B 4x16 f32 : lane L<16 -> N=L, {v0,v1}={K0,K1}; lane>=16 -> N=L-16, {K2,K3}
//   C/D 16x16  : VGPR v -> M = v + (lane>=16 ? 8 : 0), N = lane & 15
// ---------------------------------------------------------------------------
template<int JT, int JMAX, int OS>
__global__ __launch_bounds__(256) void gemm_wmma_kernel(
    const float* __restrict__ in, const float* __restrict__ W,
    const float* __restrict__ bias, float* __restrict__ out)
{
    const int wave  = threadIdx.x >> 5;
    const int lane  = threadIdx.x & 31;
    const int strip = blockIdx.x * 8 + wave;
    if (strip >= N_NODES / 16) return;   // uniform per wave

    const int m0    = strip * 16;
    const int mlane = m0 + (lane & 15);
    const int koff  = (lane >> 4) * 2;   // 0 for lanes 0-15, 2 for lanes 16-31

    // Preload all 16 A fragments (float2 per lane per k-step).
    v2f a[16];
#pragma unroll
    for (int kk = 0; kk < 16; ++kk)
        a[kk] = *(const v2f*)(in + mlane * 64 + kk * 4 + koff);

#pragma unroll
    for (int jt = 0; jt < JT; ++jt) {
        const int j = jt * 16 + (lane & 15);
        const float bv = (j < JMAX) ? bias[j] : 0.0f;
        v8f c = {bv, bv, bv, bv, bv, bv, bv, bv};
#pragma unroll
        for (int kk = 0; kk < 16; ++kk) {
            v2f b = {0.0f, 0.0f};
            if (j < JMAX)
                b = *(const v2f*)(W + j * 64 + kk * 4 + koff);
            c = __builtin_amdgcn_wmma_f32_16x16x4_f32(
                    /*neg_a=*/false, a[kk], /*neg_b=*/false, b,
                    /*c_mod=*/(short)0, c, /*reuse_a=*/false, /*reuse_b=*/false);
        }
        const int rbase = m0 + (lane >> 4) * 8;
        const int cbase = jt * 16 + (lane & 15);
#pragma unroll
        for (int v = 0; v < 8; ++v)
            out[(long long)(rbase + v) * OS + cbase] = c[v];
    }
}

// ---------------------------------------------------------------------------
// Edge scatter: agg[col[e], :] += gate[e] * h[row[e], :]
// GROUPS threads per edge, each owning a float4 channel group.
// ---------------------------------------------------------------------------
template<int GROUPS, int STRIDE>
__global__ __launch_bounds__(256) void edge_scatter_kernel(
    const float* __restrict__ h, const int* __restrict__ row,
    const int* __restrict__ col, const float* __restrict__ gate,
    float* __restrict__ agg)
{
    const int gid = blockIdx.x * blockDim.x + threadIdx.x;
    if (gid >= E_EDGES * GROUPS) return;
    const int e = gid / GROUPS;
    const int g = gid % GROUPS;
    const int r = row[e];
    const int c = col[e];
    const float gt = gate[e];
    const float4 v = *(const float4*)(h + (long long)r * STRIDE + g * 4);
    float* dst = agg + (long long)c * STRIDE + g * 4;
    atomicAdd(dst + 0, gt * v.x);
    atomicAdd(dst + 1, gt * v.y);
    atomicAdd(dst + 2, gt * v.z);
    atomicAdd(dst + 3, gt * v.w);
}

// ---------------------------------------------------------------------------
// LayerNorm over 64 channels (+optional residual, optional ReLU).
// One wave per row, 2 elems/lane, shfl_xor tree reductions (wave32).
// ---------------------------------------------------------------------------
__global__ __launch_bounds__(256) void ln64_kernel(
    const float* __restrict__ in, const float* __restrict__ res,
    const float* __restrict__ gamma, const float* __restrict__ beta,
    float* __restrict__ out, int relu)
{
    const int wave = threadIdx.x >> 5;
    const int lane = threadIdx.x & 31;
    const int r = blockIdx.x * 8 + wave;
    if (r >= N_NODES) return;
    const long long base = (long long)r * 64;
    float x0 = in[base + lane];
    float x1 = in[base + lane + 32];
    if (res) { x0 += res[base + lane]; x1 += res[base + lane + 32]; }
    float s = x0 + x1;
#pragma unroll
    for (int m = 16; m > 0; m >>= 1) s += __shfl_xor(s, m, 32);
    const float mean = s * (1.0f / 64.0f);
    const float d0 = x0 - mean, d1 = x1 - mean;
    float vs = d0 * d0 + d1 * d1;
#pragma unroll
    for (int m = 16; m > 0; m >>= 1) vs += __shfl_xor(vs, m, 32);
    const float inv = rsqrtf(vs * (1.0f / 64.0f) + 1e-5f);
    float y0 = d0 * inv * gamma[lane]      + beta[lane];
    float y1 = d1 * inv * gamma[lane + 32] + beta[lane + 32];
    if (relu) { y0 = fmaxf(y0, 0.0f); y1 = fmaxf(y1, 0.0f); }
    out[base + lane]      = y0;
    out[base + lane + 32] = y1;
}

// ---------------------------------------------------------------------------
// Final LayerNorm over 153 channels. Reads padded stride-160 buffer,
// writes compact stride-153 output. One wave per row, 5 elems/lane.
// ---------------------------------------------------------------------------
__global__ __launch_bounds__(256) void ln153_kernel(
    const float* __restrict__ in, const float* __restrict__ gamma,
    const float* __restrict__ beta, float* __restrict__ out)
{
    const int wave = threadIdx.x >> 5;
    const int lane = threadIdx.x & 31;
    const int r = blockIdx.x * 8 + wave;
    if (r >= N_NODES) return;
    const float* p = in + (long long)r * DOUT_P;
    float x[5];
    float s = 0.0f;
#pragma unroll
    for (int i = 0; i < 5; ++i) {
        const int idx = lane + 32 * i;
        x[i] = (idx < DOUT) ? p[idx] : 0.0f;
        s += x[i];
    }
#pragma unroll
    for (int m = 16; m > 0; m >>= 1) s += __shfl_xor(s, m, 32);
    const float mean = s * (1.0f / (float)DOUT);
    float vs = 0.0f;
#pragma unroll
    for (int i = 0; i < 5; ++i) {
        const int idx = lane + 32 * i;
        if (idx < DOUT) { const float d = x[i] - mean; vs += d * d; }
    }
#pragma unroll
    for (int m = 16; m > 0; m >>= 1) vs += __shfl_xor(vs, m, 32);
    const float inv = rsqrtf(vs * (1.0f / (float)DOUT) + 1e-5f);
#pragma unroll
    for (int i = 0; i < 5; ++i) {
        const int idx = lane + 32 * i;
        if (idx < DOUT)
            out[(long long)r * DOUT + idx] = (x[i] - mean) * inv * gamma[idx] + beta[idx];
    }
}

// ---------------------------------------------------------------------------
// Launch
// ---------------------------------------------------------------------------
extern "C" void kernel_launch(void* const* d_in, const int* in_sizes, int n_in,
                              void* d_out, int out_size, void* d_ws, size_t ws_size,
                              hipStream_t stream)
{
    const float* x      = (const float*)d_in[0];
    const int*   ei     = (const int*)  d_in[1];   // [2, E]
    const float* trust  = (const float*)d_in[2];
    const float* pcoef  = (const float*)d_in[3];
    const float* W1     = (const float*)d_in[4];
    const float* b1     = (const float*)d_in[5];
    const float* W2     = (const float*)d_in[6];
    const float* b2     = (const float*)d_in[7];
    const float* W3     = (const float*)d_in[8];
    const float* b3     = (const float*)d_in[9];
    const float* g1     = (const float*)d_in[10];
    const float* be1    = (const float*)d_in[11];
    const float* g2     = (const float*)d_in[12];
    const float* be2    = (const float*)d_in[13];
    const float* g3     = (const float*)d_in[14];
    const float* be3    = (const float*)d_in[15];

    const int* row = ei;
    const int* col = ei + E_EDGES;

    // Workspace carve-up (floats).
    float* ws    = (float*)d_ws;
    float* gate  = ws;                          size_t off = E_EDGES;
    float* bufP  = ws + off;  off += (size_t)N_NODES * HDIM;    // pre-activations L1/L2
    float* bufG  = ws + off;  off += (size_t)N_NODES * HDIM;    // agg L1/L2 (then h2 in place)
    float* bufH1 = ws + off;  off += (size_t)N_NODES * HDIM;    // h1 (residual)
    float* bufP3 = ws + off;  off += (size_t)N_NODES * DOUT_P;  // layer-3 pre-activation
    float* bufG3 = ws + off;  off += (size_t)N_NODES * DOUT_P;  // layer-3 agg

    const int nStrips   = N_NODES / 16;               // 6250 (exact)
    const int gemmGrid  = (nStrips + 7) / 8;          // 8 waves / block
    const int lnGrid    = (N_NODES + 7) / 8;          // 8 rows / block
    const int zero64N   = N_NODES * HDIM;
    const int zero160N  = N_NODES * DOUT_P;
    const int scat16Gr  = (E_EDGES * 16 + 255) / 256;
    const int scat40Gr  = (E_EDGES * 40 + 255) / 256;

    // Edge gate (shared by all 3 layers).
    edge_gate_kernel<<<(E_EDGES + 255) / 256, 256, 0, stream>>>(row, trust, pcoef, gate);

    // ----- Layer 1 -----
    gemm_wmma_kernel<4, HDIM, HDIM><<<gemmGrid, 256, 0, stream>>>(x, W1, b1, bufP);
    fill_zero_kernel<<<(zero64N + 255) / 256, 256, 0, stream>>>(bufG, zero64N);
    edge_scatter_kernel<16, HDIM><<<scat16Gr, 256, 0, stream>>>(bufP, row, col, gate, bufG);
    ln64_kernel<<<lnGrid, 256, 0, stream>>>(bufG, nullptr, g1, be1, bufH1, 1);

    // ----- Layer 2 (residual with h1) -----
    gemm_wmma_kernel<4, HDIM, HDIM><<<gemmGrid, 256, 0, stream>>>(bufH1, W2, b2, bufP);
    fill_zero_kernel<<<(zero64N + 255) / 256, 256, 0, stream>>>(bufG, zero64N);
    edge_scatter_kernel<16, HDIM><<<scat16Gr, 256, 0, stream>>>(bufP, row, col, gate, bufG);
    ln64_kernel<<<lnGrid, 256, 0, stream>>>(bufG, bufH1, g2, be2, bufG, 1);  // h2 in place

    // ----- Layer 3 (DOUT=153, padded stride 160; pad cols are exact 0) -----
    gemm_wmma_kernel<10, DOUT, DOUT_P><<<gemmGrid, 256, 0, stream>>>(bufG, W3, b3, bufP3);
    fill_zero_kernel<<<(zero160N + 255) / 256, 256, 0, stream>>>(bufG3, zero160N);
    edge_scatter_kernel<40, DOUT_P><<<scat40Gr, 256, 0, stream>>>(bufP3, row, col, gate, bufG3);
    ln153_kernel<<<lnGrid, 256, 0, stream>>>(bufG3, g3, be3, (float*)d_out);
}